// SpikeLoss_47021301957067
// MI455X (gfx1250) — compile-verified
//
#include <hip/hip_runtime.h>
#include <stdint.h>

typedef uint32_t u32x4 __attribute__((ext_vector_type(4)));
typedef uint32_t u32x8 __attribute__((ext_vector_type(8)));
typedef float    v8f   __attribute__((ext_vector_type(8)));
typedef float    v2f   __attribute__((ext_vector_type(2)));

#define T_STEPS 64
#define N_DIM   256
#define B_DIM   4096
#define BN      (B_DIM * N_DIM)        // 1,048,576 elements per t-slice
#define COLS    1024                   // 4 b-rows * 256 n per block
#define TC      2                      // t-steps per chunk
#define NC      (T_STEPS / TC)         // 32 chunks
#define BUF_FLOATS (2 * TC * COLS)     // out+tar tile per parity (4096 f32)

// ---------------------------------------------------------------------------
// Issue one 2D TDM tile load: tile_h rows of tile_w f32 elements, rows
// row_stride_elems apart in global memory, packed contiguously into LDS.
// Descriptor per CDNA5 ISA ch.8 (D# group0: 4 SGPRs, group1: 8 SGPRs).
// tensor_dim == tile_dim so no OOB clipping is ever triggered.
// ---------------------------------------------------------------------------
__device__ __forceinline__ void tdm_load_tile(uint32_t lds_byte_off,
                                              const float* gptr,
                                              uint32_t tile_w, uint32_t tile_h,
                                              uint32_t row_stride_elems) {
  uint64_t ga = (uint64_t)(uintptr_t)gptr;
  u32x4 g0;
  g0[0] = 0x1u;                                    // count=1 (valid user D#)
  g0[1] = lds_byte_off;                            // lds_addr [63:32]
  g0[2] = (uint32_t)ga;                            // global_addr[31:0]
  g0[3] = (uint32_t)((ga >> 32) & 0x01FFFFFFu)     // global_addr[56:32]
        | (2u << 30);                              // type=2 ("image")
  u32x8 g1;
  g1[0] = (2u << 16);                              // data_size=2 -> 4 bytes
  g1[1] = (tile_w & 0xFFFFu) << 16;                // tensor_dim0[15:0]
  g1[2] = ((tile_w >> 16) & 0xFFFFu)               // tensor_dim0[31:16]
        | ((tile_h & 0xFFFFu) << 16);              // tensor_dim1[15:0]
  g1[3] = ((tile_h >> 16) & 0xFFFFu)               // tensor_dim1[31:16]
        | ((tile_w & 0xFFFFu) << 16);              // tile_dim0
  g1[4] = (tile_h & 0xFFFFu);                      // tile_dim1 (tile_dim2=0)
  g1[5] = row_stride_elems;                        // tensor_dim0_stride[31:0]
  g1[6] = 0u;                                      // stride hi / dim1_stride lo
  g1[7] = 0u;
  asm volatile("tensor_load_to_lds %0, %1" :: "s"(g0), "s"(g1) : "memory");
}

// ---------------------------------------------------------------------------
// Stage 1: one block = 4 batch rows x 256 neurons = 1024 spike-train rows.
// TDM double-buffers [TC x 1024] f32 tiles of both arrays into LDS; threads
// build 64-bit spike masks per row, then evaluate the timing loss via
// ctz-compaction. Deterministic per-block partial sum -> d_ws[block].
// ---------------------------------------------------------------------------
__global__ void __launch_bounds__(256)
spike_loss_stage1(const float* __restrict__ outp, const float* __restrict__ tarp,
                  const int* __restrict__ label, float* __restrict__ partial) {
  extern __shared__ float sm[];
  const int tid = threadIdx.x;
  const int blk = blockIdx.x;                      // 0..1023
  const uint32_t colBase = (uint32_t)blk * COLS;   // offset in flattened [b*N+n]
  const uint32_t smBase  = (uint32_t)(uintptr_t)(&sm[0]);  // LDS byte offset

  // Prologue: wave 0 issues chunk 0 into parity 0 (2 DMAs: out, tar).
  if (tid < 32) {
    tdm_load_tile(smBase,                 outp + colBase, COLS, TC, BN);
    tdm_load_tile(smBase + TC * COLS * 4, tarp + colBase, COLS, TC, BN);
  }

  uint64_t mo[4] = {0, 0, 0, 0};
  uint64_t mt[4] = {0, 0, 0, 0};

  for (int c = 0; c < NC; ++c) {
    if (tid < 32) {
      if (c + 1 < NC) {
        const uint32_t p = (uint32_t)(c + 1) & 1u;
        const uint32_t pbase = smBase + p * (BUF_FLOATS * 4);
        const float* go = outp + (size_t)(c + 1) * TC * BN + colBase;
        const float* gt = tarp + (size_t)(c + 1) * TC * BN + colBase;
        tdm_load_tile(pbase,                 go, COLS, TC, BN);
        tdm_load_tile(pbase + TC * COLS * 4, gt, COLS, TC, BN);
        __builtin_amdgcn_s_wait_tensorcnt(2);      // chunk c complete
      } else {
        __builtin_amdgcn_s_wait_tensorcnt(0);
      }
    }
    __syncthreads();                               // publish chunk c
    const uint32_t p = (uint32_t)c & 1u;
    const float* bo = sm + p * BUF_FLOATS;
    const float* bt = bo + TC * COLS;
#pragma unroll
    for (int tl = 0; tl < TC; ++tl) {
      const float4 o = *(const float4*)(bo + tl * COLS + 4 * tid);
      const float4 g = *(const float4*)(bt + tl * COLS + 4 * tid);
      const int t = c * TC + tl;
      mo[0] |= (uint64_t)(o.x > 0.5f) << t;
      mo[1] |= (uint64_t)(o.y > 0.5f) << t;
      mo[2] |= (uint64_t)(o.z > 0.5f) << t;
      mo[3] |= (uint64_t)(o.w > 0.5f) << t;
      mt[0] |= (uint64_t)(g.x > 0.5f) << t;
      mt[1] |= (uint64_t)(g.y > 0.5f) << t;
      mt[2] |= (uint64_t)(g.z > 0.5f) << t;
      mt[3] |= (uint64_t)(g.w > 0.5f) << t;
    }
    __syncthreads();                               // done reading parity p
  }

  // Evaluate loss from the masks. k-th sorted value = ctz of mask (+1),
  // padding value = T+1 = 65. Tail (both exhausted): delta = +0.5/T, kept
  // only on the label row.
  const int b_local = tid >> 6;
  const int n0 = (tid & 63) << 2;
  const int lab = label[4 * blk + b_local];
  const float invT = 1.0f / 64.0f;
  float acc = 0.0f;
#pragma unroll
  for (int j = 0; j < 4; ++j) {
    const bool correct = (n0 + j) == lab;
    uint64_t a = mo[j], b = mt[j];
    float local = 0.0f;
    int k = 0;
    while (a | b) {
      const int ot = a ? (__builtin_ctzll(a) + 1) : 65;
      const int tt = b ? (__builtin_ctzll(b) + 1) : 65;
      a &= a - 1;                                  // 0 stays 0
      b &= b - 1;
      const float d = ((float)(ot - tt) + 0.5f) * invT;
      const bool keep = correct ? (d >= 0.0f) : (d <= 0.0f);
      local += keep ? d * d : 0.0f;
      ++k;
    }
    if (correct) {
      const float pad = 0.5f * invT;
      local += (float)(64 - k) * pad * pad;
    }
    acc += local;
  }

  // wave32 reduction, then 8 wave partials through LDS (deterministic order).
#pragma unroll
  for (int off = 16; off > 0; off >>= 1) acc += __shfl_xor(acc, off, 32);
  if ((tid & 31) == 0) sm[tid >> 5] = acc;
  __syncthreads();
  if (tid == 0) {
    float s = 0.0f;
#pragma unroll
    for (int w = 0; w < 8; ++w) s += sm[w];
    partial[blk] = s;
  }
}

// ---------------------------------------------------------------------------
// Stage 2: sum the 1024 block partials with chained V_WMMA_F32_16X16X4_F32
// (A = 16x4 partials tile, B = ones). f32 WMMA is exact and deterministic.
// D[m][0] lives in lane 0 (M=0..7) and lane 16 (M=8..15), VGPRs 0..7.
// ---------------------------------------------------------------------------
__global__ void __launch_bounds__(32)
spike_loss_stage2(const float* __restrict__ partial, float* __restrict__ outv) {
  const int lane = threadIdx.x;                    // 0..31, one wave
  v8f c = {};
  const v2f ones = {1.0f, 1.0f};
  const int m = lane & 15;
  const int koff = (lane < 16) ? 0 : 2;            // A 16x4 f32 VGPR layout
#pragma unroll
  for (int i = 0; i < 16; ++i) {
    const float* base = partial + i * 64;
    v2f a;
    a.x = base[m * 4 + koff + 0];
    a.y = base[m * 4 + koff + 1];
    c = __builtin_amdgcn_wmma_f32_16x16x4_f32(false, a, false, ones,
                                              (short)0, c, false, false);
  }
  float s = c[0] + c[1] + c[2] + c[3] + c[4] + c[5] + c[6] + c[7];
  const float tot = __shfl(s, 0, 32) + __shfl(s, 16, 32);
  if (lane == 0) outv[0] = 0.5f * tot;
}

// ---------------------------------------------------------------------------
extern "C" void kernel_launch(void* const* d_in, const int* in_sizes, int n_in,
                              void* d_out, int out_size, void* d_ws, size_t ws_size,
                              hipStream_t stream) {
  const float* outp  = (const float*)d_in[0];      // [T,B,N] f32
  const float* tarp  = (const float*)d_in[1];      // [T,B,N] f32
  const int*   label = (const int*)d_in[2];        // [B] i32
  float* partial = (float*)d_ws;                   // 1024 f32 partials
  const size_t shmem = (size_t)(2 * BUF_FLOATS) * sizeof(float);  // 32 KB
  spike_loss_stage1<<<B_DIM / 4, 256, shmem, stream>>>(outp, tarp, label, partial);
  spike_loss_stage2<<<1, 32, 0, stream>>>(partial, (float*)d_out);
}